// Evolution_19112604467873
// MI455X (gfx1250) — compile-verified
//
#include <hip/hip_runtime.h>
#include <hip/hip_bf16.h>
#include <math.h>

// Problem dims (from reference): T=32, B=16, N=2048, D=64
#define TT   32
#define BB   16
#define NN   2048
#define DD   64
#define RR   (BB * NN)          // 32768 rows per step
#define BNDS ((size_t)RR * DD)  // floats per [B,N,D] slab

// LDS row stride for 64-col tiles: 68 = 64+4 keeps float4 (16B) alignment
// (68 % 4 == 0) and spreads the K-strided A-fragment reads across banks
// (68 % 64 == 4 -> 16 consecutive rows hit 16 distinct banks).
#define LSTR 68

typedef __attribute__((ext_vector_type(2))) float v2f;
typedef __attribute__((ext_vector_type(4))) float v4f;
typedef __attribute__((ext_vector_type(8))) float v8f;
typedef __attribute__((ext_vector_type(4))) int   v4i;

#if defined(__AMDGCN__) && __has_builtin(__builtin_amdgcn_global_load_async_to_lds_b128)
#define HAVE_ASYNC_LDS 1
// src param is a GENERIC v4i* (per hipcc diagnostic) -> plain cast, const dropped.
#define SRCV4I(p) ((v4i*)(p))
// dst param: LDS as(3) v4i*. A generic pointer to LDS carries the LDS byte
// offset in its low 32 bits, so
// integer truncation yields the 32-bit as(3) pointer.
#define LDSV4I(p) ((__attribute__((address_space(3))) v4i*)(unsigned int)(unsigned long long)(p))
#else
#define HAVE_ASYNC_LDS 0
#endif

static __device__ __forceinline__ void wait_async_then_barrier() {
#if defined(__AMDGCN__)
#if __has_builtin(__builtin_amdgcn_s_wait_asynccnt)
    __builtin_amdgcn_s_wait_asynccnt(0);
#else
    asm volatile("s_wait_asynccnt 0" ::: "memory");
#endif
#endif
    __syncthreads();
}

// One recurrence step, fused:  z = [h|s] @ w1 ;  h' = sigmoid(z*g + h*(1-g))
// Grid: R/64 blocks of 128 threads (4 wave32s). Each wave computes a
// 16(M) x 64(N) output stripe via V_WMMA_F32_16X16X4_F32, K accumulated
// over 128 (64 from h, 64 from s) = 32 WMMA per 16x16 tile.
__global__ __launch_bounds__(128) void evo_step_kernel(
    const float* __restrict__ h,     // [R,64]  previous hidden (all_dyn[t-1])
    const float* __restrict__ s,     // [R,64]  static feats for this step
    const float* __restrict__ gate,  // [R]     thre_nc for this step
    const float* __restrict__ w1,    // [128,64] row-major
    float* __restrict__ hnew,        // [R,64]  all_dyn[t]
    float* __restrict__ diff)        // [R,64]  diffs[t-1]
{
    __shared__ float sh_h[64 * LSTR];
    __shared__ float sh_s[64 * LSTR];
    __shared__ float sh_g[64];

    const int    tid     = threadIdx.x;
    const size_t rowbase = (size_t)blockIdx.x * 64;

    // ---- Stage 64x64 h and s tiles into LDS ----
    // CDNA5 path: GLOBAL_LOAD_ASYNC_TO_LDS_B128 (ASYNCcnt), bypassing VGPRs.
    const float* hg = h + rowbase * DD;
    const float* sg = s + rowbase * DD;
#pragma unroll
    for (int i = 0; i < 8; ++i) {
        int linear = (i * 128 + tid) * 4;      // 0..4092, step 4; covers 64*64
        int r = linear >> 6;                   // row 0..63
        int c = linear & 63;                   // col 0..63 (multiple of 4)
#if HAVE_ASYNC_LDS
        __builtin_amdgcn_global_load_async_to_lds_b128(
            SRCV4I(hg + linear), LDSV4I(&sh_h[r * LSTR + c]), 0, 0);
        __builtin_amdgcn_global_load_async_to_lds_b128(
            SRCV4I(sg + linear), LDSV4I(&sh_s[r * LSTR + c]), 0, 0);
#else
        *(v4f*)(&sh_h[r * LSTR + c]) = *(const v4f*)(hg + linear);
        *(v4f*)(&sh_s[r * LSTR + c]) = *(const v4f*)(sg + linear);
#endif
    }
    if (tid < 64) sh_g[tid] = gate[rowbase + tid];
    wait_async_then_barrier();

    // ---- Per-wave WMMA compute ----
    const int w    = tid >> 5;       // wave id 0..3 -> rows [16w,16w+16)
    const int lane = tid & 31;
    const int hi   = lane >> 4;      // lane half selects K pair / row half
    const int lo   = lane & 15;
    const int arow = w * 16 + lo;    // A-matrix row this lane supplies

#pragma unroll
    for (int ct = 0; ct < 4; ++ct) {             // 4 x 16 output columns
        v8f acc = {0.f, 0.f, 0.f, 0.f, 0.f, 0.f, 0.f, 0.f};
#pragma unroll
        for (int p = 0; p < 2; ++p) {            // p=0: h half, p=1: s half
            const float* xs = p ? sh_s : sh_h;
#pragma unroll
            for (int kk = 0; kk < 16; ++kk) {    // K = 64 per half, 4 per WMMA
                const int kb = kk * 4 + hi * 2;  // this lane-half's K pair
                // A 16x4 fp32: lanes0-15 K{0,1}, lanes16-31 K{2,3} (ISA 7.12.2)
                v2f a;
                a.x = xs[arow * LSTR + kb];
                a.y = xs[arow * LSTR + kb + 1];
                // B 4x16 fp32: row=K, col striped over lanes (mirror of A)
                v2f b;
                b.x = w1[(size_t)(p * 64 + kb)     * DD + ct * 16 + lo];
                b.y = w1[(size_t)(p * 64 + kb + 1) * DD + ct * 16 + lo];
                acc = __builtin_amdgcn_wmma_f32_16x16x4_f32(
                    /*neg_a=*/false, a, /*neg_b=*/false, b,
                    /*c_mod=*/(short)0, acc, /*reuse_a=*/false, /*reuse_b=*/false);
            }
        }
        // ---- Epilogue: gate + sigmoid + stores ----
#pragma unroll
        for (int v = 0; v < 8; ++v) {            // C/D: VGPR v -> M = v + 8*hi
            const int   rl   = w * 16 + v + 8 * hi;
            const int   c    = ct * 16 + lo;
            const float hold = sh_h[rl * LSTR + c];
            const float g    = sh_g[rl];
            const float x    = acc[v] * g + hold * (1.0f - g);
            const float hn   = 1.0f / (1.0f + __expf(-x));
            const size_t gi  = (rowbase + rl) * DD + c;
            hnew[gi] = hn;
            diff[gi] = hn - hold;
        }
    }
}

extern "C" void kernel_launch(void* const* d_in, const int* in_sizes, int n_in,
                              void* d_out, int out_size, void* d_ws, size_t ws_size,
                              hipStream_t stream) {
    (void)in_sizes; (void)n_in; (void)out_size; (void)d_ws; (void)ws_size;

    const float* all_static = (const float*)d_in[0];  // [T,B,N,D]
    const float* thre       = (const float*)d_in[1];  // [T,B,N,1]
    const float* h0         = (const float*)d_in[2];  // [B,N,D]
    const float* w1         = (const float*)d_in[3];  // [128,64]

    float* out     = (float*)d_out;
    float* all_dyn = out;                        // [T,B,N,D]
    float* finalp  = out + (size_t)TT * BNDS;    // [B,N,D]
    float* diffs   = finalp + BNDS;              // [T-1,B,N,D]

    // all_dyn[0] = initial hidden state
    (void)hipMemcpyAsync(all_dyn, h0, BNDS * sizeof(float),
                         hipMemcpyDeviceToDevice, stream);

    dim3 grid(RR / 64), block(128);
    for (int t = 1; t < TT; ++t) {
        evo_step_kernel<<<grid, block, 0, stream>>>(
            all_dyn + (size_t)(t - 1) * BNDS,          // h (prev step output)
            all_static + (size_t)t * BNDS,             // s_t
            thre + (size_t)t * RR,                     // gate_t
            w1,
            all_dyn + (size_t)t * BNDS,                // h_new
            diffs + (size_t)(t - 1) * BNDS);           // diff
    }

    // final = all_dyn[T-1]
    (void)hipMemcpyAsync(finalp, all_dyn + (size_t)(TT - 1) * BNDS,
                         BNDS * sizeof(float), hipMemcpyDeviceToDevice, stream);
}